// SparseConv3D_33749853012468
// MI455X (gfx1250) — compile-verified
//
#include <hip/hip_runtime.h>

typedef float v2f __attribute__((ext_vector_type(2)));
typedef float v8f __attribute__((ext_vector_type(8)));

#define CIN   64
#define COUT  64
#define NTAPS 27
#define WSTRIDE (NTAPS * CIN)   // 1728 floats per cout slice of weight
#define CENTER_TAP 13           // (1,1,1) in 3x3x3, dz-major

// ---------------------------------------------------------------------------
// Center tap: out[n][co] = bias[co] + sum_k sp[n][k] * W[co][center][k]
// One block = 16 voxel rows; 4 waves cover the 4 cout tiles of 16.
// K=64 reduction = 16 x V_WMMA_F32_16X16X4_F32 (exact fp32 path).
// ---------------------------------------------------------------------------
__global__ __launch_bounds__(128) void sparse_conv_center(
    const float* __restrict__ sp, const float* __restrict__ weight,
    const float* __restrict__ bias, float* __restrict__ out, int N)
{
    const int lane     = threadIdx.x & 31;
    const int wave     = threadIdx.x >> 5;       // 0..3 -> cout tile
    const int row0     = blockIdx.x * 16;
    const int m        = lane & 15;              // A row / B col / D col role
    const int khalf    = lane >> 4;
    const int coutBase = wave * 16;

    // A fragment source: row = row0 + m (clamped, masked)
    const int  row   = row0 + m;
    const float vmask = (row < N) ? 1.0f : 0.0f;
    const int  rowc  = (row < N) ? row : 0;
    const float* aBase = sp + (size_t)rowc * CIN + 2 * khalf;

    // B fragment source: W[co= coutBase+m][center tap][k], k contiguous
    const float* bBase = weight + (size_t)(coutBase + m) * WSTRIDE
                                + CENTER_TAP * CIN + 2 * khalf;

    v8f c = {};
    #pragma unroll
    for (int k = 0; k < CIN; k += 4) {
        v2f a = *(const v2f*)(aBase + k);        // A[m][k+2*khalf .. +1]
        a.x *= vmask; a.y *= vmask;
        v2f b = *(const v2f*)(bBase + k);        // B[k+2*khalf .. +1][m]
        c = __builtin_amdgcn_wmma_f32_16x16x4_f32(
                /*neg_a=*/false, a, /*neg_b=*/false, b,
                /*c_mod=*/(short)0, c, /*reuse_a=*/false, /*reuse_b=*/false);
    }

    // D layout: VGPR r -> row r + 8*(lane>>4), col lane&15
    const float bv = bias[coutBase + m];
    #pragma unroll
    for (int r = 0; r < 8; ++r) {
        const int rr = row0 + r + 8 * khalf;
        if (rr < N)
            out[(size_t)rr * COUT + coutBase + m] = c[r] + bv;
    }
}

// ---------------------------------------------------------------------------
// 26 neighbor taps: gather 16 pairs -> 16x64 @ 64x64 -> atomic scatter-add.
// grid.x = pair tiles, grid.y = offset o in [0,26). Tiles past sizes[o]
// early-exit uniformly (EXEC stays all-1s for every WMMA).
// ---------------------------------------------------------------------------
__global__ __launch_bounds__(128) void sparse_conv_offsets(
    const float* __restrict__ sp, const float* __restrict__ weight,
    const int* __restrict__ nei, const int* __restrict__ sizes,
    float* __restrict__ out, int P)
{
    const int o  = blockIdx.y;
    const int tb = blockIdx.x * 16;
    const int sz = sizes[o];
    if (tb >= sz) return;                         // wave-uniform exit

    const int lane     = threadIdx.x & 31;
    const int wave     = threadIdx.x >> 5;
    const int coutBase = wave * 16;
    const int m        = lane & 15;
    const int khalf    = lane >> 4;

    const int tap = (o < CENTER_TAP) ? o : o + 1; // skip center in 0..26

    // Gather row for A fragment
    const int   p     = tb + m;
    const float vmask = (p < sz) ? 1.0f : 0.0f;
    const int   pc    = (p < P) ? p : (P - 1);
    const int   inIdx = nei[(o * P + pc) * 2 + 1];
    const float* aBase = sp + (size_t)inIdx * CIN + 2 * khalf;
    const float* bBase = weight + (size_t)(coutBase + m) * WSTRIDE
                                + tap * CIN + 2 * khalf;

    v8f c = {};
    #pragma unroll
    for (int k = 0; k < CIN; k += 4) {
        v2f a = *(const v2f*)(aBase + k);
        a.x *= vmask; a.y *= vmask;               // zero padded pair rows
        v2f b = *(const v2f*)(bBase + k);
        c = __builtin_amdgcn_wmma_f32_16x16x4_f32(
                false, a, false, b, (short)0, c, false, false);
    }

    // Scatter-add: row r+8*khalf of the tile -> out[nei[o][pr][0]]
    #pragma unroll
    for (int r = 0; r < 8; ++r) {
        const int pr = tb + r + 8 * khalf;
        if (pr < sz) {
            const int outRow = nei[(o * P + pr) * 2 + 0];
            atomicAdd(&out[(size_t)outRow * COUT + coutBase + m], c[r]);
        }
    }
}

// ---------------------------------------------------------------------------
extern "C" void kernel_launch(void* const* d_in, const int* in_sizes, int n_in,
                              void* d_out, int out_size, void* d_ws, size_t ws_size,
                              hipStream_t stream) {
    const float* sp     = (const float*)d_in[0];   // [N,64]
    const float* weight = (const float*)d_in[1];   // [64,3,3,3,64]
    const float* bias   = (const float*)d_in[2];   // [64]
    const int*   nei    = (const int*)d_in[3];     // [26,P,2]
    const int*   sizes  = (const int*)d_in[4];     // [26]
    float* out = (float*)d_out;

    const int N = in_sizes[0] / CIN;
    const int P = in_sizes[3] / (26 * 2);

    const int rowTiles = (N + 15) / 16;
    sparse_conv_center<<<rowTiles, 128, 0, stream>>>(sp, weight, bias, out, N);

    const int pairTiles = (P + 15) / 16;
    dim3 grid(pairTiles, 26);
    sparse_conv_offsets<<<grid, 128, 0, stream>>>(sp, weight, nei, sizes, out, P);
}